// MemoryEfficientGNN_17781164606118
// MI455X (gfx1250) — compile-verified
//
#include <hip/hip_runtime.h>
#include <hip/hip_bf16.h>
#include <math.h>

// ---- problem constants (match reference) ----
#define IN_DIM      128
#define POS_DIM     16
#define CONTENT_DIM 112          // IN - POS
#define H_HEADS     4
#define OUT_DIM     16
#define HO          64           // H * OUT
#define NEG_SLOPE   0.2f
#define DIV_W       0.1f

typedef float v2f __attribute__((ext_vector_type(2)));
typedef float v8f __attribute__((ext_vector_type(8)));

// Order-preserving float<->uint map so we can use integer atomicMax for the
// segment-softmax max (handles negative scores correctly).
__device__ __forceinline__ unsigned enc_f32(float f) {
    unsigned u = __float_as_uint(f);
    return (u & 0x80000000u) ? ~u : (u | 0x80000000u);
}
__device__ __forceinline__ float dec_f32(unsigned u) {
    unsigned b = (u & 0x80000000u) ? (u & 0x7FFFFFFFu) : ~u;
    return __uint_as_float(b);
}

// ---------------------------------------------------------------------------
// K1: content = x[:, :112] @ W   via V_WMMA_F32_16X16X4_F32  (f32, K=4)
// One wave computes a 16-node x 64-col tile; block of 8 waves = 128 nodes.
// A-frag layout (ISA 7.12.2, 32-bit A 16x4): lanes 0-15 hold K=0(v0),K=1(v1),
// lanes 16-31 hold K=2(v0),K=3(v1), M = lane&15. B-frag mirrors (N = lane&15).
//
// W is staged in LDS K-pair interleaved:  lds[(k>>1)*128 + c*2 + (k&1)]
// so each B fragment {W[kb][c], W[kb+1][c]} (kb even) is one aligned
// ds_load_b64 straight into the even-aligned VGPR pair WMMA consumes.
// ---------------------------------------------------------------------------
__global__ __launch_bounds__(256)
void k_content_gemm(const float* __restrict__ x, const float* __restrict__ W,
                    float* __restrict__ content, int n)
{
    __shared__ float lds_w[CONTENT_DIM * HO];   // 28 KB of the 320 KB WGP LDS
    // pair-interleaved fill: thread handles (k-pair, col) entries
    for (int i = threadIdx.x; i < (CONTENT_DIM / 2) * HO; i += 256) {
        const int kp = i >> 6;                  // k pair index (k = 2*kp)
        const int c  = i & 63;
        v2f wv = { W[(2 * kp) * HO + c], W[(2 * kp + 1) * HO + c] };
        *(v2f*)&lds_w[kp * 128 + c * 2] = wv;
    }
    __syncthreads();

    const int wave = threadIdx.x >> 5;          // wave32
    const int lane = threadIdx.x & 31;
    const int node_base = blockIdx.x * 128 + wave * 16;

    const int m  = lane & 15;                   // M (A) / N (B,C) within tile
    const int kk = (lane >> 4) * 2;             // lane-half selects K pair

    int arow = node_base + m;
    if (arow >= n) arow = n - 1;                // clamp; stores masked below
    const float* __restrict__ xr = x + (long)arow * IN_DIM;

    v8f acc0 = {}, acc1 = {}, acc2 = {}, acc3 = {};

    for (int k = 0; k < CONTENT_DIM; k += 4) {
        const int kb = k + kk;                  // even
        v2f a = *(const v2f*)(xr + kb);         // A[m][kb], A[m][kb+1]
        const float* wp = &lds_w[(kb >> 1) * 128];
        v2f b0 = *(const v2f*)(wp + ( 0 + m) * 2);
        v2f b1 = *(const v2f*)(wp + (16 + m) * 2);
        v2f b2 = *(const v2f*)(wp + (32 + m) * 2);
        v2f b3 = *(const v2f*)(wp + (48 + m) * 2);
        acc0 = __builtin_amdgcn_wmma_f32_16x16x4_f32(false, a, false, b0, (short)0, acc0, false, false);
        acc1 = __builtin_amdgcn_wmma_f32_16x16x4_f32(false, a, false, b1, (short)0, acc1, false, false);
        acc2 = __builtin_amdgcn_wmma_f32_16x16x4_f32(false, a, false, b2, (short)0, acc2, false, false);
        acc3 = __builtin_amdgcn_wmma_f32_16x16x4_f32(false, a, false, b3, (short)0, acc3, false, false);
    }

    // C/D layout: VGPR v, lanes 0-15 -> M=v, lanes 16-31 -> M=v+8, N=lane&15
    const int mbase = (lane >> 4) * 8;
#pragma unroll
    for (int v = 0; v < 8; ++v) {
        const int node = node_base + mbase + v;
        if (node < n) {
            float* cr = content + (long)node * HO;
            cr[ 0 + m] = acc0[v];
            cr[16 + m] = acc1[v];
            cr[32 + m] = acc2[v];
            cr[48 + m] = acc3[v];
        }
    }
}

// ---------------------------------------------------------------------------
// K2: init out rows with bias, zero softmax accumulators and gram
// ---------------------------------------------------------------------------
__global__ void k_init(float* __restrict__ out, const float* __restrict__ bias,
                       unsigned* __restrict__ maxenc, float* __restrict__ sumb,
                       float* __restrict__ gram, int n)
{
    const long total = (long)n * HO;
    const long t = (long)blockIdx.x * blockDim.x + threadIdx.x;
    if (t < total) out[t] = bias[t & (HO - 1)];
    if (t < (long)n * H_HEADS) { maxenc[t] = 0u; sumb[t] = 0.0f; }  // 0 == -inf key
    if (t < 16) gram[t] = 0.0f;
}

// ---------------------------------------------------------------------------
// K3: per-(node,head) attention source/target scores
// ---------------------------------------------------------------------------
__global__ void k_scores(const float* __restrict__ x, const float* __restrict__ content,
                         const float* __restrict__ att, const float* __restrict__ pos_att,
                         float* __restrict__ s_i, float* __restrict__ s_j, int n)
{
    const int t = blockIdx.x * blockDim.x + threadIdx.x;
    if (t >= n * H_HEADS) return;
    const int node = t >> 2, h = t & 3;

    const float* c  = content + (long)node * HO + h * OUT_DIM;
    const float* ai = att + h * 2 * OUT_DIM;
    const float* aj = ai + OUT_DIM;
    float si = 0.f, sj = 0.f;
#pragma unroll
    for (int d = 0; d < OUT_DIM; ++d) { si += c[d] * ai[d]; sj += c[d] * aj[d]; }

    const float* p  = x + (long)node * IN_DIM + CONTENT_DIM;
    const float* pi = pos_att + h * 2 * POS_DIM;
    const float* pj = pi + POS_DIM;
#pragma unroll
    for (int d = 0; d < POS_DIM; ++d) { si += p[d] * pi[d]; sj += p[d] * pj[d]; }

    s_i[t] = si; s_j[t] = sj;
}

__device__ __forceinline__ void edge_rc(const int* __restrict__ ei, int idx, int e,
                                        int& r, int& c)
{
    if (idx < e) { r = ei[idx]; c = ei[e + idx]; }
    else         { r = c = idx - e; }            // appended self-loops
}

// ---------------------------------------------------------------------------
// K4: segment max over `row` (uint-encoded float atomicMax)
// ---------------------------------------------------------------------------
__global__ void k_edge_max(const int* __restrict__ ei, const float* __restrict__ s_i,
                           const float* __restrict__ s_j, unsigned* __restrict__ maxenc,
                           int n, int e)
{
    const long t = (long)blockIdx.x * blockDim.x + threadIdx.x;
    if (t >= (long)(e + n) * H_HEADS) return;
    const int idx = (int)(t >> 2), h = (int)(t & 3);
    int r, c; edge_rc(ei, idx, e, r, c);
    float s = s_i[r * 4 + h] + s_j[c * 4 + h];
    s = (s > 0.f) ? s : s * NEG_SLOPE;
    atomicMax(&maxenc[r * 4 + h], enc_f32(s));
}

// ---------------------------------------------------------------------------
// K5: segment sum of exp(score - max) over `row`  (score recomputed, no Et buffer)
// ---------------------------------------------------------------------------
__global__ void k_edge_sum(const int* __restrict__ ei, const float* __restrict__ s_i,
                           const float* __restrict__ s_j, const unsigned* __restrict__ maxenc,
                           float* __restrict__ sumb, int n, int e)
{
    const long t = (long)blockIdx.x * blockDim.x + threadIdx.x;
    if (t >= (long)(e + n) * H_HEADS) return;
    const int idx = (int)(t >> 2), h = (int)(t & 3);
    int r, c; edge_rc(ei, idx, e, r, c);
    float s = s_i[r * 4 + h] + s_j[c * 4 + h];
    s = (s > 0.f) ? s : s * NEG_SLOPE;
    const float ex = expf(s - dec_f32(maxenc[r * 4 + h]));
    atomicAdd(&sumb[r * 4 + h], ex);
}

// ---------------------------------------------------------------------------
// K6: normalize, scatter messages into out (col), accumulate gram (wave-shuffle
//     reduce -> LDS -> 16 global atomics per block)
// ---------------------------------------------------------------------------
__global__ __launch_bounds__(256)
void k_edge_scatter(const int* __restrict__ ei, const float* __restrict__ s_i,
                    const float* __restrict__ s_j, const unsigned* __restrict__ maxenc,
                    const float* __restrict__ sumb, const float* __restrict__ content,
                    float* __restrict__ out, float* __restrict__ gram, int n, int e)
{
    __shared__ float lgram[16];
    if (threadIdx.x < 16) lgram[threadIdx.x] = 0.0f;
    __syncthreads();

    const long t = (long)blockIdx.x * blockDim.x + threadIdx.x;
    const int  lane = threadIdx.x & 31;
    float al[H_HEADS] = {0.f, 0.f, 0.f, 0.f};

    if (t < (long)(e + n)) {
        int r, c; edge_rc(ei, (int)t, e, r, c);
#pragma unroll
        for (int h = 0; h < H_HEADS; ++h) {
            float s = s_i[r * 4 + h] + s_j[c * 4 + h];
            s = (s > 0.f) ? s : s * NEG_SLOPE;
            const float ex = expf(s - dec_f32(maxenc[r * 4 + h]));
            al[h] = ex / (sumb[r * 4 + h] + 1e-10f);
        }
        // message scatter: out[col] += content[row] * alpha  (resident in L2)
        const float4* __restrict__ cr = (const float4*)(content + (long)r * HO);
        float* __restrict__ orow = out + (long)c * HO;
#pragma unroll
        for (int h = 0; h < H_HEADS; ++h) {
            const float a = al[h];
#pragma unroll
            for (int q = 0; q < 4; ++q) {
                const float4 v = cr[h * 4 + q];
                atomicAdd(&orow[h * 16 + q * 4 + 0], v.x * a);
                atomicAdd(&orow[h * 16 + q * 4 + 1], v.y * a);
                atomicAdd(&orow[h * 16 + q * 4 + 2], v.z * a);
                atomicAdd(&orow[h * 16 + q * 4 + 3], v.w * a);
            }
        }
    }

    // gram[h1][h2] += alpha[h1]*alpha[h2] : reduce across the wave first
    float p[16];
#pragma unroll
    for (int h1 = 0; h1 < 4; ++h1)
#pragma unroll
        for (int h2 = 0; h2 < 4; ++h2)
            p[h1 * 4 + h2] = al[h1] * al[h2];
#pragma unroll
    for (int off = 16; off >= 1; off >>= 1)
#pragma unroll
        for (int i = 0; i < 16; ++i)
            p[i] += __shfl_down(p[i], off, 32);
    if (lane == 0)
#pragma unroll
        for (int i = 0; i < 16; ++i) atomicAdd(&lgram[i], p[i]);
    __syncthreads();
    if (threadIdx.x < 16) atomicAdd(&gram[threadIdx.x], lgram[threadIdx.x]);
}

// ---------------------------------------------------------------------------
// K7: diversity loss from the 4x4 gram
// ---------------------------------------------------------------------------
__global__ void k_finalize(const float* __restrict__ gram, float* __restrict__ loss)
{
    if (blockIdx.x == 0 && threadIdx.x == 0) {
        float nrm[H_HEADS];
#pragma unroll
        for (int h = 0; h < H_HEADS; ++h)
            nrm[h] = fmaxf(sqrtf(gram[h * 4 + h]), 1e-12f);
        float acc = 0.f;
#pragma unroll
        for (int h1 = 0; h1 < H_HEADS; ++h1)
#pragma unroll
            for (int h2 = 0; h2 < H_HEADS; ++h2)
                if (h1 != h2) acc += gram[h1 * 4 + h2] / (nrm[h1] * nrm[h2]);
        loss[0] = (acc / 16.0f) * DIV_W;     // mean over HxH, diag zeroed
    }
}

extern "C" void kernel_launch(void* const* d_in, const int* in_sizes, int n_in,
                              void* d_out, int out_size, void* d_ws, size_t ws_size,
                              hipStream_t stream)
{
    const float* x       = (const float*)d_in[0];
    const int*   ei      = (const int*)  d_in[1];   // (2, E) row-major, int32
    const float* W       = (const float*)d_in[2];
    const float* att     = (const float*)d_in[3];
    const float* pos_att = (const float*)d_in[4];
    const float* bias    = (const float*)d_in[5];
    float*       out     = (float*)d_out;           // [N*64] out, then [1] loss

    const int n = in_sizes[0] / IN_DIM;
    const int e = in_sizes[1] / 2;

    // workspace layout (all re-initialized every call)
    float*    content = (float*)d_ws;                       // n*64
    float*    s_i     = content + (size_t)n * HO;           // n*4
    float*    s_j     = s_i + (size_t)n * H_HEADS;          // n*4
    unsigned* maxenc  = (unsigned*)(s_j + (size_t)n * H_HEADS); // n*4
    float*    sumb    = (float*)(maxenc + (size_t)n * H_HEADS); // n*4
    float*    gram    = sumb + (size_t)n * H_HEADS;         // 16

    const int threads = 256;
    const int eh = (e + n) * H_HEADS;

    k_content_gemm<<<(n + 127) / 128, threads, 0, stream>>>(x, W, content, n);
    k_init<<<((long)n * HO + threads - 1) / threads, threads, 0, stream>>>(
        out, bias, maxenc, sumb, gram, n);
    k_scores<<<(n * H_HEADS + threads - 1) / threads, threads, 0, stream>>>(
        x, content, att, pos_att, s_i, s_j, n);
    k_edge_max<<<(eh + threads - 1) / threads, threads, 0, stream>>>(
        ei, s_i, s_j, maxenc, n, e);
    k_edge_sum<<<(eh + threads - 1) / threads, threads, 0, stream>>>(
        ei, s_i, s_j, maxenc, sumb, n, e);
    k_edge_scatter<<<(e + n + threads - 1) / threads, threads, 0, stream>>>(
        ei, s_i, s_j, maxenc, sumb, content, out, gram, n, e);
    k_finalize<<<1, 32, 0, stream>>>(gram, out + (size_t)n * HO);
}